// RRF_Online_9698036154531
// MI455X (gfx1250) — compile-verified
//
#include <hip/hip_runtime.h>

// Online RRF SGD: strictly sequential scan over T steps.
// Persistent 1-block / 4-wave kernel; eps held in VGPRs in WMMA operand
// layout; per-step matvecs via V_WMMA_F32_16X16X4_F32 accumulate chains.

#define T_STEPS 200000
#define NFEAT   128
#define NSPEC   64
#define LRW     0.05f
#define LRG     0.05f

typedef __attribute__((ext_vector_type(2))) float v2f;
typedef __attribute__((ext_vector_type(8))) float v8f;

__device__ __forceinline__ v8f wmma_f32(v2f a, v2f b, v8f c) {
  // 8-arg pattern: (neg_a, A, neg_b, B, c_mod, C, reuse_a, reuse_b)
  return __builtin_amdgcn_wmma_f32_16x16x4_f32(
      false, a, false, b, (short)0, c, false, false);
}

__launch_bounds__(128, 1)
__global__ void rrf_online_kernel(const float* __restrict__ X,
                                  const float* __restrict__ Y,
                                  const float* __restrict__ gamma0,
                                  const float* __restrict__ w0,
                                  const float* __restrict__ eps,
                                  float* __restrict__ preds) {
  __shared__ float xeLds[NFEAT];     // xe = x * exp(gamma), feature-major
  __shared__ float comboLds[NSPEC];  // err*(cos*w_hi - sin*w_lo)
  __shared__ float termLds[NFEAT];   // eps @ combo
  __shared__ __align__(16) float predPart[4];

  const int tid  = threadIdx.x;
  const int wt   = tid >> 5;     // wave 0..3
  const int lane = tid & 31;
  const int lo   = lane & 15;
  const int hi   = lane >> 4;    // 0 or 1 (half-wave)
  const int f    = wt * 32 + lane;   // this thread's feature index

  // ---------------- resident state -------------------------------------
  float gamma_r = gamma0[f];                 // gamma[f]
  float wA = w0[wt * 16 + lo];               // w[n],       n = 16*wt + lo
  float wB = w0[NSPEC + wt * 16 + lo];       // w[64 + n]

  // ---------------- eps packed into WMMA operand registers -------------
  // mv1: phi_x(1x64) = xe(1x128) @ eps(128x64). Wave wt owns N-tile
  // columns [16wt, 16wt+16). B chunk c covers K(features) 4c..4c+3.
  // B f32 4x16 layout: lane n<16: (B[0,n],B[1,n]); lane 16+n: (B[2,n],B[3,n]).
  v2f b1[32];
#pragma unroll
  for (int c = 0; c < 32; ++c) {
    int k0 = 4 * c + 2 * hi;
    v2f b;
    b.x = eps[(k0    ) * NSPEC + wt * 16 + lo];
    b.y = eps[(k0 + 1) * NSPEC + wt * 16 + lo];
    b1[c] = b;
  }
  // mv2: term(128) = eps(128x64) @ combo(64). Wave wt owns M-tiles
  // 2wt, 2wt+1 (rows 32wt..32wt+31). A f32 16x4 layout: lane m<16:
  // (A[m,0],A[m,1]); lane 16+m: (A[m,2],A[m,3]).
  v2f a2[2][16];
#pragma unroll
  for (int mt = 0; mt < 2; ++mt) {
#pragma unroll
    for (int c = 0; c < 16; ++c) {
      int row = 16 * (2 * wt + mt) + lo;
      int c0  = 4 * c + 2 * hi;
      v2f a;
      a.x = eps[row * NSPEC + c0];
      a.y = eps[row * NSPEC + c0 + 1];
      a2[mt][c] = a;
    }
  }

  // ---------------- X/Y software pipeline (depth 4) --------------------
  float x0 = X[0 * NFEAT + f], x1 = X[1 * NFEAT + f];
  float x2 = X[2 * NFEAT + f], x3 = X[3 * NFEAT + f];
  float y0 = Y[0], y1 = Y[1], y2 = Y[2], y3 = Y[3];

  __syncthreads();

  for (int t = 0; t < T_STEPS; ++t) {
    float x = x0, y = y0;
    x0 = x1; x1 = x2; x2 = x3;
    y0 = y1; y1 = y2; y2 = y3;
    int tn = (t + 4 < T_STEPS) ? (t + 4) : (T_STEPS - 1);
    x3 = X[(size_t)tn * NFEAT + f];
    y3 = Y[tn];
    if (t + 64 < T_STEPS)  // warm L2 well ahead (global_prefetch_b8)
      __builtin_prefetch(X + (size_t)(t + 64) * NFEAT + lane * 4, 0, 1);

    // xe = x * exp(gamma)
    float eg = __expf(gamma_r);
    float xe = x * eg;
    xeLds[f] = xe;
    __syncthreads();  // (1) xe visible to all waves

    // ---- mv1: phi_x tile via 32 chained WMMAs (row 0 meaningful) ----
    v8f acc = {};
#pragma unroll
    for (int c = 0; c < 32; ++c) {
      // lane0: xe[4c],xe[4c+1]; lane16: xe[4c+2],xe[4c+3]; others dup (finite)
      v2f a = *(const v2f*)((const char*)xeLds + c * 16 + hi * 8);
      acc = wmma_f32(a, b1[c], acc);
    }
    float phi = acc[0];          // lanes 0-15: phi_x[16wt + lo]
    float cn = __cosf(phi);
    float sn = __sinf(phi);

    // ---- prediction: dot(phi, w), 16-lane tree + cross-wave via LDS ----
    float p = cn * wA + sn * wB;       // valid on lanes 0-15
    p += __shfl_xor(p, 1);
    p += __shfl_xor(p, 2);
    p += __shfl_xor(p, 4);
    p += __shfl_xor(p, 8);
    if (lane == 0) predPart[wt] = p;
    __syncthreads();  // (2) partials visible

    float4 pp = *(const float4*)predPart;
    float pred = (pp.x + pp.y) + (pp.z + pp.w);
    float err  = pred - y;
    bool  nan  = !(pred == pred);

    if (wt == 0 && lane == 0) preds[t] = pred;

    // combo uses OLD w (d_phi = err * w before the update)
    float combo = err * (cn * wB - sn * wA);
    if (lane < 16) comboLds[wt * 16 + lo] = combo;

    // w update: w -= lr*(lr*exp(w) + err*phi), skipped on NaN pred
    float wAn = wA - LRW * (LRW * __expf(wA) + err * cn);
    float wBn = wB - LRW * (LRW * __expf(wB) + err * sn);
    wA = nan ? wA : wAn;
    wB = nan ? wB : wBn;
    __syncthreads();  // (3) combo visible

    // ---- mv2: term slice via 2 x 16 chained WMMAs (col 0 meaningful) ----
#pragma unroll
    for (int mt = 0; mt < 2; ++mt) {
      v8f acc2 = {};
#pragma unroll
      for (int c = 0; c < 16; ++c) {
        v2f b = *(const v2f*)((const char*)comboLds + c * 16 + hi * 8);
        acc2 = wmma_f32(a2[mt][c], b, acc2);
      }
      // column 0 lives in lanes 0 (rows r) and 16 (rows 8+r)
#pragma unroll
      for (int r = 0; r < 8; ++r) {
        if (lo == 0) termLds[32 * wt + 16 * mt + hi * 8 + r] = acc2[r];
      }
    }
    // termLds[f] is produced by this same wave -> wave-internal DS ordering
    float term = termLds[f];
    float gn = gamma_r - LRG * (x * term * eg);
    gamma_r = nan ? gamma_r : gn;
    // next iteration's xe store is fenced from this iteration's reads by
    // barriers (2)/(3) already passed by every wave -> no extra barrier
  }
}

extern "C" void kernel_launch(void* const* d_in, const int* in_sizes, int n_in,
                              void* d_out, int out_size, void* d_ws, size_t ws_size,
                              hipStream_t stream) {
  (void)in_sizes; (void)n_in; (void)out_size; (void)d_ws; (void)ws_size;
  const float* X      = (const float*)d_in[0];
  const float* Y      = (const float*)d_in[1];
  const float* gamma0 = (const float*)d_in[2];
  const float* w0     = (const float*)d_in[3];
  const float* eps    = (const float*)d_in[4];
  float* preds        = (float*)d_out;
  rrf_online_kernel<<<1, 128, 0, stream>>>(X, Y, gamma0, w0, eps, preds);
}